// AbsoluteRelativePositionEmbedding_70549132804760
// MI455X (gfx1250) — compile-verified
//
#include <hip/hip_runtime.h>
#include <hip/hip_bf16.h>

// ---------------------------------------------------------------------------
// Problem constants (from setup_inputs in the reference)
// ---------------------------------------------------------------------------
#define B_SZ   4
#define N_PTS  8192
#define K_NN   20
#define NK_TOT (N_PTS * K_NN)      // 163840 positions per batch
#define EPS_GN 1e-5f

#if __has_builtin(__builtin_amdgcn_wmma_f32_16x16x4_f32)
#define HAS_WMMA_F32X4 1
#else
#define HAS_WMMA_F32X4 0
#endif

typedef __attribute__((ext_vector_type(2))) float v2f;
typedef __attribute__((ext_vector_type(8))) float v8f;

__device__ __forceinline__ float eluf(float x) { return x > 0.f ? x : expm1f(x); }

// order-preserving float<->uint mapping for exact atomic max
__device__ __forceinline__ unsigned encf(float f) {
  unsigned u = __float_as_uint(f);
  return (u & 0x80000000u) ? ~u : (u | 0x80000000u);
}
__device__ __forceinline__ float decf(unsigned u) {
  return __uint_as_float((u & 0x80000000u) ? (u & 0x7fffffffu) : ~u);
}

// ---------------------------------------------------------------------------
// Kernel 1: KNN (WMMA f32 16x16x4 inner products) + build x0 = [center, nbr-center]
// Each wave owns 16 query rows (one per lane%16); one V_WMMA_F32_16X16X4_F32
// per 16-candidate tile. ALL 32 lanes do selection: lane half 0 scans tile
// columns [0,8), half 1 scans [8,16), each with a private sorted top-20;
// the two half-lists are merged stably (tie -> lower index) at the end.
// x0 layout: [B][6][N][K]
// ---------------------------------------------------------------------------
__global__ void __launch_bounds__(256) knn_x0_kernel(const float* __restrict__ pts,
                                                     float* __restrict__ x0) {
  __shared__ float sInner[8][16][17];       // per-wave WMMA tile, padded (bank-conflict free)
  __shared__ float sSQC[16];                // candidate squared norms for current tile
  __shared__ float mD[8][16][2][K_NN];      // per-wave half-lists for final merge
  __shared__ int   mI[8][16][2][K_NN];
#if !HAS_WMMA_F32X4
  __shared__ float sX[16], sY[16], sZ[16];
#endif

  const int tid  = threadIdx.x;
  const int wv   = tid >> 5;
  const int lane = tid & 31;
  const int half = lane >> 4;
  const int l15  = lane & 15;

  const int b   = blockIdx.x / (N_PTS / 128);
  const int qb0 = (blockIdx.x % (N_PTS / 128)) * 128;
  const float* Pb = pts + (size_t)b * 3 * N_PTS;

  const int qa = qb0 + wv * 16 + l15;   // query row this lane holds (A reg + selection)

  // A matrix regs: 16x4 (K padded 3->4 with zero)
  v2f a;
  a.x = half ? Pb[2 * N_PTS + qa] : Pb[qa];
  a.y = half ? 0.f               : Pb[N_PTS + qa];

  // per-lane query state (both halves own row l15 of the tile)
  const float cx = Pb[qa], cy = Pb[N_PTS + qa], cz = Pb[2 * N_PTS + qa];
  const float sqq = cx * cx + cy * cy + cz * cz;

  float bestd[K_NN];
  int   besti[K_NN];
#pragma unroll
  for (int s = 0; s < K_NN; ++s) { bestd[s] = 3.0e38f; besti[s] = 0; }

  for (int jb = 0; jb < N_PTS; jb += 16) {
    __syncthreads();  // WAR protection for sSQC / sInner reads of previous tile

    // B matrix regs: 4x16 (K padded)
    v2f bm;
    bm.x = half ? Pb[2 * N_PTS + jb + l15] : Pb[jb + l15];
    bm.y = half ? 0.f                      : Pb[N_PTS + jb + l15];

    if (wv == 0) {
      // low lanes hold (x,y) of candidate l15, high lanes hold z -> combine via shfl
      const float zz = __shfl(bm.x, lane + 16, 32);   // lanes<16: z of candidate l15
      if (lane < 16) sSQC[lane] = fmaf(zz, zz, fmaf(bm.x, bm.x, bm.y * bm.y));
#if !HAS_WMMA_F32X4
      if (half) sZ[l15] = bm.x;
      else      { sX[l15] = bm.x; sY[l15] = bm.y; }
#endif
    }

#if HAS_WMMA_F32X4
    v8f cacc = {};
    cacc = __builtin_amdgcn_wmma_f32_16x16x4_f32(false, a, false, bm,
                                                 (short)0, cacc, false, false);
#pragma unroll
    for (int v = 0; v < 8; ++v)
      sInner[wv][v + 8 * half][l15] = cacc[v];   // C layout: vgpr v -> M=v(+8), N=lane%16
#endif
    __syncthreads();

    // selection: all 32 lanes active; half h scans columns [8h, 8h+8)
#pragma unroll
    for (int nn = 0; nn < 8; ++nn) {
      const int n = half * 8 + nn;
#if HAS_WMMA_F32X4
      const float inner = sInner[wv][l15][n];
#else
      const float inner = cx * sX[n] + cy * sY[n] + cz * sZ[n];
#endif
      const float d = sqq + sSQC[n] - 2.0f * inner;
      const int j = jb + n;
      if (d < bestd[K_NN - 1]) {
        // fully-unrolled register insertion (stable: ties keep lower index first)
#pragma unroll
        for (int s = K_NN - 1; s > 0; --s) {
          if (bestd[s] > d) {
            const bool shift = bestd[s - 1] > d;
            bestd[s] = shift ? bestd[s - 1] : d;
            besti[s] = shift ? besti[s - 1] : j;
          }
        }
        if (bestd[0] > d) { bestd[0] = d; besti[0] = j; }
      }
    }
  }

  // dump both half-lists, then lanes 0..15 merge (stable) and emit features
#pragma unroll
  for (int s = 0; s < K_NN; ++s) { mD[wv][l15][half][s] = bestd[s]; mI[wv][l15][half][s] = besti[s]; }
  __syncthreads();

  if (lane < 16) {
    const size_t NK = (size_t)NK_TOT;
    float* xb = x0 + (size_t)b * 6 * NK + (size_t)qa * K_NN;
    int ia = 0, ib = 0;
    for (int s = 0; s < K_NN; ++s) {
      const float dA = mD[wv][lane][0][ia]; const int jA = mI[wv][lane][0][ia];
      const float dB = mD[wv][lane][1][ib]; const int jB = mI[wv][lane][1][ib];
      const bool takeA = (dA < dB) || (dA == dB && jA < jB);
      const int j = takeA ? jA : jB;
      if (takeA) ++ia; else ++ib;
      const float nx = Pb[j], ny = Pb[N_PTS + j], nz = Pb[2 * N_PTS + j];
      xb[0 * NK + s] = cx;
      xb[1 * NK + s] = cy;
      xb[2 * NK + s] = cz;
      xb[3 * NK + s] = nx - cx;
      xb[4 * NK + s] = ny - cy;
      xb[5 * NK + s] = nz - cz;
    }
  }
}

// ---------------------------------------------------------------------------
// Kernel 2: conv1 (6->32) pre-GN stats: per-block partial sum/sumsq per group.
// partial layout: part[blockIdx.x*16 + s], s<8 = group sums, s>=8 = group sumsqs
// ---------------------------------------------------------------------------
__global__ void __launch_bounds__(256) conv1_stats_kernel(const float* __restrict__ x0,
                                                          const float* __restrict__ w1,
                                                          const float* __restrict__ b1,
                                                          float* __restrict__ part) {
  __shared__ float sW[32 * 6], sB[32];
  __shared__ float warr[8][16];
  const int tid = threadIdx.x;
  if (tid < 192) sW[tid] = w1[tid];
  if (tid < 32)  sB[tid] = b1[tid];
  __syncthreads();

  const int bpb = NK_TOT / 256;               // 640 blocks per batch
  const int b = blockIdx.x / bpb;
  const int p = (blockIdx.x % bpb) * 256 + tid;

  float xin[6];
#pragma unroll
  for (int c = 0; c < 6; ++c) xin[c] = x0[((size_t)b * 6 + c) * NK_TOT + p];

  float gs[8], gss[8];
#pragma unroll
  for (int g = 0; g < 8; ++g) {
    float s = 0.f, ss = 0.f;
#pragma unroll
    for (int cc = 0; cc < 4; ++cc) {
      const int c = g * 4 + cc;
      float acc = sB[c];
#pragma unroll
      for (int ci = 0; ci < 6; ++ci) acc = fmaf(sW[c * 6 + ci], xin[ci], acc);
      s += acc; ss += acc * acc;
    }
    gs[g] = s; gss[g] = ss;
  }

  const int lane = tid & 31, wv = tid >> 5;
#pragma unroll
  for (int g = 0; g < 8; ++g) {
    float s = gs[g], q = gss[g];
#pragma unroll
    for (int o = 16; o > 0; o >>= 1) { s += __shfl_down(s, o, 32); q += __shfl_down(q, o, 32); }
    if (lane == 0) { warr[wv][g] = s; warr[wv][8 + g] = q; }
  }
  __syncthreads();
  if (tid < 16) {
    float s = 0.f;
#pragma unroll
    for (int w = 0; w < 8; ++w) s += warr[w][tid];
    part[(size_t)blockIdx.x * 16 + tid] = s;
  }
}

// Deterministic fixed-order reduction of per-block partials -> stats[B*16]
__global__ void reduce_stats_kernel(const float* __restrict__ part,
                                    float* __restrict__ stats, int nblk) {
  const int t = threadIdx.x;            // 64 threads: b*16 + s
  const int b = t >> 4, r = t & 15;
  float s = 0.f;
  for (int i = 0; i < nblk; ++i) s += part[(size_t)(b * nblk + i) * 16 + r];
  stats[t] = s;
}

// ---------------------------------------------------------------------------
// Shared helper: recompute x1 = ELU(GN1(conv1(x0))) for one position (b,p)
// ---------------------------------------------------------------------------
__device__ __forceinline__ void compute_x1(const float* __restrict__ x0,
                                           const float* sW1, const float* sB1,
                                           const float* sG1, const float* sBe1,
                                           const float* __restrict__ S1,
                                           int b, int p, float* x1) {
  float xin[6];
#pragma unroll
  for (int c = 0; c < 6; ++c) xin[c] = x0[((size_t)b * 6 + c) * NK_TOT + p];
#pragma unroll
  for (int c = 0; c < 32; ++c) {
    float acc = sB1[c];
#pragma unroll
    for (int ci = 0; ci < 6; ++ci) acc = fmaf(sW1[c * 6 + ci], xin[ci], acc);
    x1[c] = acc;
  }
  const float inv = 1.0f / (4.0f * (float)NK_TOT);   // group count = (32/8)*N*K
#pragma unroll
  for (int g = 0; g < 8; ++g) {
    const float m = S1[b * 16 + g] * inv;
    const float v = S1[b * 16 + 8 + g] * inv - m * m;
    const float r = rsqrtf(v + EPS_GN);
#pragma unroll
    for (int cc = 0; cc < 4; ++cc) {
      const int c = g * 4 + cc;
      x1[c] = eluf((x1[c] - m) * r * sG1[c] + sBe1[c]);
    }
  }
}

// ---------------------------------------------------------------------------
// Kernel 3: conv2 (32->64) pre-GN stats (recomputes x1 on the fly)
// ---------------------------------------------------------------------------
__global__ void __launch_bounds__(256) conv2_stats_kernel(const float* __restrict__ x0,
                                                          const float* __restrict__ w1,
                                                          const float* __restrict__ b1,
                                                          const float* __restrict__ g1,
                                                          const float* __restrict__ be1,
                                                          const float* __restrict__ S1,
                                                          const float* __restrict__ w2,
                                                          const float* __restrict__ b2,
                                                          float* __restrict__ part) {
  __shared__ float sW1[192], sB1[32], sG1[32], sBe1[32];
  __shared__ float sW2[2048], sB2[64];
  __shared__ float warr[8][16];
  const int tid = threadIdx.x;
  for (int i = tid; i < 2048; i += 256) sW2[i] = w2[i];
  if (tid < 192) sW1[tid] = w1[tid];
  if (tid < 32)  { sB1[tid] = b1[tid]; sG1[tid] = g1[tid]; sBe1[tid] = be1[tid]; }
  if (tid < 64)  sB2[tid] = b2[tid];
  __syncthreads();

  const int bpb = NK_TOT / 256;
  const int b = blockIdx.x / bpb;
  const int p = (blockIdx.x % bpb) * 256 + tid;

  float x1[32];
  compute_x1(x0, sW1, sB1, sG1, sBe1, S1, b, p, x1);

  float gs[8], gss[8];
#pragma unroll
  for (int g = 0; g < 8; ++g) {
    float s = 0.f, ss = 0.f;
    for (int cc = 0; cc < 8; ++cc) {
      const int c2 = g * 8 + cc;
      float acc = sB2[c2];
#pragma unroll
      for (int ci = 0; ci < 32; ++ci) acc = fmaf(sW2[c2 * 32 + ci], x1[ci], acc);
      s += acc; ss += acc * acc;
    }
    gs[g] = s; gss[g] = ss;
  }

  const int lane = tid & 31, wv = tid >> 5;
#pragma unroll
  for (int g = 0; g < 8; ++g) {
    float s = gs[g], q = gss[g];
#pragma unroll
    for (int o = 16; o > 0; o >>= 1) { s += __shfl_down(s, o, 32); q += __shfl_down(q, o, 32); }
    if (lane == 0) { warr[wv][g] = s; warr[wv][8 + g] = q; }
  }
  __syncthreads();
  if (tid < 16) {
    float s = 0.f;
#pragma unroll
    for (int w = 0; w < 8; ++w) s += warr[w][tid];
    part[(size_t)blockIdx.x * 16 + tid] = s;
  }
}

// ---------------------------------------------------------------------------
// Kernel 4: init + max over N of x2 = ELU(GN2(conv2(x1))); exact atomic max
// x3u layout: [B][64][K] as order-preserving uints
// ---------------------------------------------------------------------------
__global__ void x3_init_kernel(unsigned* __restrict__ x3u) {
  const int i = blockIdx.x * 256 + threadIdx.x;
  if (i < B_SZ * 64 * K_NN) x3u[i] = 0x007FFFFFu;   // enc(-inf)
}

__global__ void __launch_bounds__(256) max_over_n_kernel(const float* __restrict__ x0,
                                                         const float* __restrict__ w1,
                                                         const float* __restrict__ b1,
                                                         const float* __restrict__ g1,
                                                         const float* __restrict__ be1,
                                                         const float* __restrict__ S1,
                                                         const float* __restrict__ w2,
                                                         const float* __restrict__ b2,
                                                         const float* __restrict__ g2,
                                                         const float* __restrict__ be2,
                                                         const float* __restrict__ S2,
                                                         unsigned* __restrict__ x3u) {
  __shared__ float sW1[192], sB1[32], sG1[32], sBe1[32];
  __shared__ float sW2[2048], sB2[64], sG2[64], sBe2[64];
  __shared__ float wmax[8][64];
  const int tid = threadIdx.x;
  for (int i = tid; i < 2048; i += 256) sW2[i] = w2[i];
  if (tid < 192) sW1[tid] = w1[tid];
  if (tid < 32)  { sB1[tid] = b1[tid]; sG1[tid] = g1[tid]; sBe1[tid] = be1[tid]; }
  if (tid < 64)  { sB2[tid] = b2[tid]; sG2[tid] = g2[tid]; sBe2[tid] = be2[tid]; }
  __syncthreads();

  const int NCH = N_PTS / 256;        // 32 n-chunks
  int rem = blockIdx.x;
  const int ch = rem % NCH; rem /= NCH;
  const int kk = rem % K_NN; rem /= K_NN;
  const int b = rem;
  const int n = ch * 256 + tid;
  const int p = n * K_NN + kk;

  float x1[32];
  compute_x1(x0, sW1, sB1, sG1, sBe1, S1, b, p, x1);

  const float inv2 = 1.0f / (8.0f * (float)NK_TOT);  // group count = (64/8)*N*K
  float m2[8], r2[8];
#pragma unroll
  for (int g = 0; g < 8; ++g) {
    const float m = S2[b * 16 + g] * inv2;
    const float v = S2[b * 16 + 8 + g] * inv2 - m * m;
    m2[g] = m; r2[g] = rsqrtf(v + EPS_GN);
  }

  const int lane = tid & 31, wv = tid >> 5;
#pragma unroll
  for (int g = 0; g < 8; ++g) {
    for (int cc = 0; cc < 8; ++cc) {
      const int c2 = g * 8 + cc;
      float acc = sB2[c2];
#pragma unroll
      for (int ci = 0; ci < 32; ++ci) acc = fmaf(sW2[c2 * 32 + ci], x1[ci], acc);
      float v = eluf((acc - m2[g]) * r2[g] * sG2[c2] + sBe2[c2]);
#pragma unroll
      for (int o = 16; o > 0; o >>= 1) v = fmaxf(v, __shfl_down(v, o, 32));
      if (lane == 0) wmax[wv][c2] = v;
    }
  }
  __syncthreads();
  if (tid < 64) {
    float m = wmax[0][tid];
#pragma unroll
    for (int w = 1; w < 8; ++w) m = fmaxf(m, wmax[w][tid]);
    atomicMax(&x3u[((size_t)b * 64 + tid) * K_NN + kk], encf(m));
  }
}

// ---------------------------------------------------------------------------
// Kernel 5: head — conv3(64->512)+GN+ELU in LDS, conv4(512->1024)+GN+ELU.
// One block per batch; GN stats are block-local (wave-per-group shfl reduce).
// ---------------------------------------------------------------------------
__global__ void __launch_bounds__(256) head_kernel(const unsigned* __restrict__ x3u,
                                                   const float* __restrict__ w3,
                                                   const float* __restrict__ b3,
                                                   const float* __restrict__ g3,
                                                   const float* __restrict__ be3,
                                                   const float* __restrict__ w4,
                                                   const float* __restrict__ b4,
                                                   const float* __restrict__ g4,
                                                   const float* __restrict__ be4,
                                                   float* __restrict__ y4,
                                                   float* __restrict__ out) {
  __shared__ float sx3[64 * K_NN];        // 1280
  __shared__ float sy3[512 * K_NN];       // 10240 (40KB)
  __shared__ float sm3[8], sr3[8], sm4[8], sr4[8];
  const int tid = threadIdx.x;
  const int b = blockIdx.x;
  const int lane = tid & 31, wv = tid >> 5;

  for (int i = tid; i < 64 * K_NN; i += 256) sx3[i] = decf(x3u[(size_t)b * 64 * K_NN + i]);
  __syncthreads();

  // conv3: y3[c][k]
  for (int o = tid; o < 512 * K_NN; o += 256) {
    const int c = o / K_NN, k = o - c * K_NN;
    float acc = b3[c];
#pragma unroll 8
    for (int c0 = 0; c0 < 64; ++c0) acc = fmaf(w3[c * 64 + c0], sx3[c0 * K_NN + k], acc);
    sy3[o] = acc;
  }
  __syncthreads();

  // GN3 stats: wave wv handles group wv (64 ch * 20 = 1280 elems)
  {
    float s = 0.f, ss = 0.f;
    const int base = wv * 64 * K_NN;
    for (int e = lane; e < 64 * K_NN; e += 32) { const float v = sy3[base + e]; s += v; ss += v * v; }
#pragma unroll
    for (int o = 16; o > 0; o >>= 1) { s += __shfl_down(s, o, 32); ss += __shfl_down(ss, o, 32); }
    if (lane == 0) {
      const float cnt = 64.0f * K_NN;
      const float m = s / cnt, v = ss / cnt - m * m;
      sm3[wv] = m; sr3[wv] = rsqrtf(v + EPS_GN);
    }
  }
  __syncthreads();
  for (int o = tid; o < 512 * K_NN; o += 256) {
    const int c = o / K_NN, g = c >> 6;
    sy3[o] = eluf((sy3[o] - sm3[g]) * sr3[g] * g3[c] + be3[c]);
  }
  __syncthreads();

  // conv4: y4[c2][k] -> global scratch (too big for remaining LDS budget)
  float* y4b = y4 + (size_t)b * 1024 * K_NN;
  for (int o = tid; o < 1024 * K_NN; o += 256) {
    const int c2 = o / K_NN, k = o - c2 * K_NN;
    float acc = b4[c2];
#pragma unroll 8
    for (int c = 0; c < 512; ++c) acc = fmaf(w4[c2 * 512 + c], sy3[c * K_NN + k], acc);
    y4b[o] = acc;
  }
  __syncthreads();

  // GN4 stats: wave wv handles group wv (128 ch * 20 = 2560 elems)
  {
    float s = 0.f, ss = 0.f;
    const int base = wv * 128 * K_NN;
    for (int e = lane; e < 128 * K_NN; e += 32) { const float v = y4b[base + e]; s += v; ss += v * v; }
#pragma unroll
    for (int o = 16; o > 0; o >>= 1) { s += __shfl_down(s, o, 32); ss += __shfl_down(ss, o, 32); }
    if (lane == 0) {
      const float cnt = 128.0f * K_NN;
      const float m = s / cnt, v = ss / cnt - m * m;
      sm4[wv] = m; sr4[wv] = rsqrtf(v + EPS_GN);
    }
  }
  __syncthreads();

  float* outb = out + (size_t)b * 1024 * K_NN;
  for (int o = tid; o < 1024 * K_NN; o += 256) {
    const int c2 = o / K_NN, g = c2 >> 7;
    outb[o] = eluf((y4b[o] - sm4[g]) * sr4[g] * g4[c2] + be4[c2]);
  }
}

// ---------------------------------------------------------------------------
// Launch
// ---------------------------------------------------------------------------
extern "C" void kernel_launch(void* const* d_in, const int* in_sizes, int n_in,
                              void* d_out, int out_size, void* d_ws, size_t ws_size,
                              hipStream_t stream) {
  (void)in_sizes; (void)n_in; (void)out_size; (void)ws_size;
  const float* pts = (const float*)d_in[0];
  const float* w1  = (const float*)d_in[1];
  const float* b1  = (const float*)d_in[2];
  const float* g1  = (const float*)d_in[3];
  const float* be1 = (const float*)d_in[4];
  const float* w2  = (const float*)d_in[5];
  const float* b2  = (const float*)d_in[6];
  const float* g2  = (const float*)d_in[7];
  const float* be2 = (const float*)d_in[8];
  const float* w3  = (const float*)d_in[9];
  const float* b3  = (const float*)d_in[10];
  const float* g3  = (const float*)d_in[11];
  const float* be3 = (const float*)d_in[12];
  const float* w4  = (const float*)d_in[13];
  const float* b4  = (const float*)d_in[14];
  const float* g4  = (const float*)d_in[15];
  const float* be4 = (const float*)d_in[16];

  // workspace layout (floats): everything fully re-written every call
  float* ws = (float*)d_ws;
  float*    X0  = ws;                                   // B*6*N*K   = 3,932,160
  float*    P1  = X0 + (size_t)B_SZ * 6 * NK_TOT;       // 2560*16   = 40,960
  float*    P2  = P1 + 2560 * 16;                       // 40,960
  float*    S1  = P2 + 2560 * 16;                       // 64
  float*    S2  = S1 + 64;                              // 64
  unsigned* X3U = (unsigned*)(S2 + 64);                 // B*64*K    = 5,120
  float*    Y4  = (float*)(X3U + B_SZ * 64 * K_NN);     // B*1024*K  = 81,920

  const int blocksNK = (B_SZ * NK_TOT) / 256;           // 2560

  knn_x0_kernel      <<<B_SZ * (N_PTS / 128), 256, 0, stream>>>(pts, X0);
  conv1_stats_kernel <<<blocksNK, 256, 0, stream>>>(X0, w1, b1, P1);
  reduce_stats_kernel<<<1, 64, 0, stream>>>(P1, S1, NK_TOT / 256);
  conv2_stats_kernel <<<blocksNK, 256, 0, stream>>>(X0, w1, b1, g1, be1, S1, w2, b2, P2);
  reduce_stats_kernel<<<1, 64, 0, stream>>>(P2, S2, NK_TOT / 256);
  x3_init_kernel     <<<(B_SZ * 64 * K_NN + 255) / 256, 256, 0, stream>>>(X3U);
  max_over_n_kernel  <<<B_SZ * K_NN * (N_PTS / 256), 256, 0, stream>>>(
      X0, w1, b1, g1, be1, S1, w2, b2, g2, be2, S2, X3U);
  head_kernel        <<<B_SZ, 256, 0, stream>>>(X3U, w3, b3, g3, be3, w4, b4, g4, be4,
                                                Y4, (float*)d_out);
}